// Block_Matmul_Module_7679401525982
// MI455X (gfx1250) — compile-verified
//
#include <hip/hip_runtime.h>
#include <hip/hip_bf16.h>

// Block-local matmul on gfx1250 (MI455X):
//   out[b, i*64+rr, j*64+oo] = sum_cc x[b, i*64+rr, j*64+cc] * weight[i*64+cc, j*64+oo]
// One workgroup per (b,i,j) 64x64x64 block. fp32 WMMA (V_WMMA_F32_16X16X4_F32).
// A-block staged with GLOBAL_LOAD_ASYNC_TO_LDS_B128 (ASYNCcnt path);
// W-block staged transposed through VGPRs (fp32 has no TR-load).

typedef __attribute__((ext_vector_type(2))) float v2f;
typedef __attribute__((ext_vector_type(8))) float v8f;

#define MDIM 4096
#define KDIM 4096
#define NDIM 4096
#define BLK 64
// LDS row stride in dwords: 64 + 4 so that the 16-lane, one-row-apart b64
// fragment loads hit all 64 banks with zero conflicts (bank = (68*r + k) % 64
// = (4r + k) % 64 -> halves at k and k+2 tile the full bank space).
#define LSTR 68

__global__ __launch_bounds__(256, 2)
void block_matmul_wmma_f32(const float* __restrict__ x,
                           const float* __restrict__ w,
                           float* __restrict__ out)
{
    const int j = blockIdx.x;   // output column block (also x column block)
    const int i = blockIdx.y;   // output row block    (also w row block)
    const int b = blockIdx.z;   // batch
    const int t = threadIdx.x;  // 0..255

    __shared__ float ldsA[BLK * LSTR];   // ldsA[m][k]  (row-major, padded)
    __shared__ float ldsB[BLK * LSTR];   // ldsB[n][k]  (W transposed, padded)

    // ---------------- stage A block: async global->LDS b128 copies.
    // Per-lane LDS destination bakes in the 68-dword padded stride.
    {
        const float* src = x + ((size_t)b * MDIM + (size_t)i * BLK) * KDIM + (size_t)j * BLK;
#pragma unroll
        for (int it = 0; it < 4; ++it) {
            int v   = t + it * 256;       // vec4 id 0..1023
            int row = v >> 4;             // 0..63
            int c4  = (v & 15) << 2;      // 0,4,..,60
            unsigned ldsOff = (unsigned)(size_t)&ldsA[row * LSTR + c4]; // LDS byte addr
            unsigned gOff   = (unsigned)(row * KDIM + c4) * 4u;         // byte offset
            asm volatile("global_load_async_to_lds_b128 %0, %1, %2"
                         :
                         : "v"(ldsOff), "v"(gOff), "s"(src)
                         : "memory");
        }
    }
    // ---------------- stage W block transposed: coalesced loads -> ds_store_b32 x4
    {
        const float* src = w + (size_t)i * BLK * NDIM + (size_t)j * BLK;
#pragma unroll
        for (int it = 0; it < 4; ++it) {
            int v   = t + it * 256;
            int k   = v >> 4;             // W row (contraction index)
            int c4  = (v & 15) << 2;      // W col (output n), group of 4
            float4 d = *(const float4*)(src + (size_t)k * NDIM + c4);
            ldsB[(c4 + 0) * LSTR + k] = d.x;
            ldsB[(c4 + 1) * LSTR + k] = d.y;
            ldsB[(c4 + 2) * LSTR + k] = d.z;
            ldsB[(c4 + 3) * LSTR + k] = d.w;
        }
    }
    // Wait for this wave's async copies, then workgroup barrier (which also
    // covers the ds_store traffic via dscnt).
    asm volatile("s_wait_asynccnt 0x0" ::: "memory");
    __syncthreads();

    // ---------------- per-wave: two 16x16 tiles sharing the A fragment
    const int wave = t >> 5;          // 0..7
    const int lane = t & 31;
    const int half = lane >> 4;       // 0: K={0,1}, 1: K={2,3} within each k-step
    const int lr   = lane & 15;

    const int tr  = wave & 3;         // tile row   0..3
    const int tc0 = (wave >> 2) << 1; // tile cols {tc0, tc0+1}, tc0 in {0,2}

    const float* aP  = &ldsA[(tr * 16 + lr) * LSTR + half * 2];
    const float* b0P = &ldsB[((tc0 + 0) * 16 + lr) * LSTR + half * 2];
    const float* b1P = &ldsB[((tc0 + 1) * 16 + lr) * LSTR + half * 2];

    v8f c0 = {};
    v8f c1 = {};
#pragma unroll
    for (int k0 = 0; k0 < BLK; k0 += 4) {
        v2f a   = *(const v2f*)(aP  + k0);   // ds_load_2addr_b64, conflict-free
        v2f bb0 = *(const v2f*)(b0P + k0);
        v2f bb1 = *(const v2f*)(b1P + k0);
        c0 = __builtin_amdgcn_wmma_f32_16x16x4_f32(false, a, false, bb0,
                                                   (short)0, c0, false, false);
        c1 = __builtin_amdgcn_wmma_f32_16x16x4_f32(false, a, false, bb1,
                                                   (short)0, c1, false, false);
    }

    // ---------------- store C/D (VGPR g -> M = tr*16 + g + half*8, N = lr)
    float* ob = out + ((size_t)b * MDIM + (size_t)i * BLK + tr * 16 + half * 8) * NDIM
                    + (size_t)j * BLK;
    const int n0 = (tc0 + 0) * 16 + lr;
    const int n1 = (tc0 + 1) * 16 + lr;
#pragma unroll
    for (int g = 0; g < 8; ++g) {
        ob[(size_t)g * NDIM + n0] = c0[g];
        ob[(size_t)g * NDIM + n1] = c1[g];
    }
}

extern "C" void kernel_launch(void* const* d_in, const int* in_sizes, int n_in,
                              void* d_out, int out_size, void* d_ws, size_t ws_size,
                              hipStream_t stream) {
    const float* x = (const float*)d_in[0];   // [4, 4096, 4096] fp32
    const float* w = (const float*)d_in[1];   // [4096, 4096]    fp32
    float* out = (float*)d_out;               // [4, 4096, 4096] fp32

    dim3 grid(NDIM / BLK, MDIM / BLK, 4);     // (j, i, b) = (64, 64, 4)
    block_matmul_wmma_f32<<<grid, 256, 0, stream>>>(x, w, out);
}